// H_STGCN_79654463472122
// MI455X (gfx1250) — compile-verified
//
#include <hip/hip_runtime.h>
#include <math.h>

#define B_    2048
#define V_    15
#define T_    16
#define H_    64
#define LH_   40
#define EPER_ 40
#define N_    (B_*V_)      // 30720
#define E_    (B_*EPER_)   // 81920
#define TH_   (T_*H_)      // 1024
#define EPS_  1e-5f

typedef __attribute__((ext_vector_type(2))) float v2f;
typedef __attribute__((ext_vector_type(8))) float v8f;

__device__ __forceinline__ float sigf(float x) { return 1.0f / (1.0f + __expf(-x)); }

// ---------------------------------------------------------------------------
// Generic fp32 WMMA GEMM:  C[M][NN] = A[M][K] * B  (+bias1+bias2, opt ReLU)
// B element (k,n) is at Bw[k*bsK + n*bsN]  (handles both [K][N] and [N][K]
// weight storage). NN = NTILES*16. K must be a multiple of 64, M of 128.
// One wave computes one 16-row tile across all NTILES column tiles using
// V_WMMA_F32_16X16X4_F32; weight chunk lives in LDS with +8 row padding so
// lanes 0-15 / 16-31 read disjoint bank halves.
// ---------------------------------------------------------------------------
template <int NTILES>
__launch_bounds__(256)
__global__ void gemm_f32_wmma(const float* __restrict__ A, int lda,
                              const float* __restrict__ Bw, int bsK, int bsN,
                              const float* __restrict__ bias1,
                              const float* __restrict__ bias2,
                              int relu,
                              float* __restrict__ C, int ldc,
                              int M, int K)
{
    constexpr int NN  = NTILES * 16;
    constexpr int WST = NN + 8;              // 2*WST % 64 == 16 -> conflict-free halves
    __shared__ float Wl[64 * WST];

    const int lane    = threadIdx.x & 31;
    const int wave    = threadIdx.x >> 5;
    const int rowTile = blockIdx.x * 8 + wave;
    const int m       = lane & 15;           // A row within tile / C column
    const int kb      = (lane >> 4) << 1;    // 0 or 2

    v8f acc[NTILES];
#pragma unroll
    for (int nt = 0; nt < NTILES; ++nt)
#pragma unroll
        for (int r = 0; r < 8; ++r) acc[nt][r] = 0.0f;

    for (int k0 = 0; k0 < K; k0 += 64) {
        // stage B chunk [64][NN] into LDS (coalesced over n)
        for (int idx = threadIdx.x; idx < 64 * NN; idx += 256) {
            int kk = idx / NN, nn = idx - kk * NN;
            Wl[kk * WST + nn] = Bw[(size_t)(k0 + kk) * bsK + (size_t)nn * bsN];
        }
        __syncthreads();

        if (rowTile * 16 < M) {
            const float* Ar = A + (size_t)(rowTile * 16 + m) * lda + k0;
#pragma unroll
            for (int kk = 0; kk < 64; kk += 4) {
                v2f a;
                a.x = Ar[kk + kb];
                a.y = Ar[kk + kb + 1];
#pragma unroll
                for (int nt = 0; nt < NTILES; ++nt) {
                    v2f b;
                    b.x = Wl[(kk + kb) * WST + nt * 16 + m];
                    b.y = Wl[(kk + kb + 1) * WST + nt * 16 + m];
                    acc[nt] = __builtin_amdgcn_wmma_f32_16x16x4_f32(
                        false, a, false, b, (short)0, acc[nt], false, false);
                }
            }
        }
        __syncthreads();
    }

    if (rowTile * 16 < M) {
        const int rbase = rowTile * 16 + ((lane < 16) ? 0 : 8);
        const int col   = lane & 15;
#pragma unroll
        for (int nt = 0; nt < NTILES; ++nt) {
            int   n  = nt * 16 + col;
            float bv = 0.0f;
            if (bias1) bv += bias1[n];
            if (bias2) bv += bias2[n];
#pragma unroll
            for (int r = 0; r < 8; ++r) {
                float v = acc[nt][r] + bv;
                if (relu) v = fmaxf(v, 0.0f);
                C[(size_t)(rbase + r) * ldc + n] = v;
            }
        }
    }
}

// ---------------------------------------------------------------------------
// conv_1 (1x1 conv): x[n,t,h] = data[n,0,t]*W1[h,0] + data[n,1,t]*W1[h,1] + b1[h]
// ---------------------------------------------------------------------------
__global__ void conv1_kernel(const float* __restrict__ data, const float* __restrict__ W1,
                             const float* __restrict__ b1, float* __restrict__ x)
{
    int idx = blockIdx.x * 256 + threadIdx.x;
    if (idx >= N_ * TH_) return;
    int n = idx >> 10, r = idx & 1023, t = r >> 6, h = r & 63;
    float d0 = data[n * 32 + t];
    float d1 = data[n * 32 + 16 + t];
    x[idx]   = d0 * W1[h * 2] + d1 * W1[h * 2 + 1] + b1[h];
}

// ---------------------------------------------------------------------------
// degree / dinv
// ---------------------------------------------------------------------------
__global__ void init_deg(float* deg)
{
    int i = blockIdx.x * 256 + threadIdx.x;
    if (i < N_) deg[i] = 1.0f;  // self loop weight
}
__global__ void acc_deg(const int* __restrict__ ei, const float* __restrict__ ew, float* deg)
{
    int e = blockIdx.x * 256 + threadIdx.x;
    if (e < E_) atomicAdd(&deg[ei[E_ + e]], ew[e]);  // col = ei[1][e]
}
__global__ void fin_dinv(float* deg)
{
    int i = blockIdx.x * 256 + threadIdx.x;
    if (i < N_) { float d = deg[i]; deg[i] = d > 0.0f ? rsqrtf(d) : 0.0f; }
}

// ---------------------------------------------------------------------------
// GCN aggregation + bias + ReLU, fused with BN statistics.
// One block = one graph x one 256-wide slice of (t,h). LDS per-node
// accumulators (each thread owns a column -> no races, no feature atomics).
// y may alias xw: every block reads exactly the region it later writes.
// ---------------------------------------------------------------------------
__launch_bounds__(256)
__global__ void gcn_agg(const float* __restrict__ xw, const int* __restrict__ ei,
                        const float* __restrict__ ew, const float* __restrict__ dinv,
                        const float* __restrict__ bias, float* __restrict__ y,
                        float* __restrict__ bnsum, float* __restrict__ bnsq)
{
    const int g  = blockIdx.x >> 2;
    const int th = ((blockIdx.x & 3) << 8) + threadIdx.x;  // 0..1023

    __shared__ float nrm[EPER_];
    __shared__ int   sl[EPER_], dl[EPER_];
    __shared__ float selfn[V_];
    __shared__ float accL[V_][256];

    if (threadIdx.x < EPER_) {
        int e = g * EPER_ + threadIdx.x;
        int s = ei[e], d = ei[E_ + e];
        sl[threadIdx.x]  = s - g * V_;
        dl[threadIdx.x]  = d - g * V_;
        nrm[threadIdx.x] = dinv[s] * ew[e] * dinv[d];
    }
    if (threadIdx.x < V_) {
        float di = dinv[g * V_ + threadIdx.x];
        selfn[threadIdx.x] = di * di;
    }
    __syncthreads();

    const float* xg = xw + (size_t)g * V_ * TH_;
#pragma unroll
    for (int v = 0; v < V_; ++v)
        accL[v][threadIdx.x] = selfn[v] * xg[v * TH_ + th];
    for (int e = 0; e < EPER_; ++e)
        accL[dl[e]][threadIdx.x] += nrm[e] * xg[sl[e] * TH_ + th];

    float*      yg = y + (size_t)g * V_ * TH_;
    const float bv = bias[th & 63];
    float s1 = 0.0f, s2 = 0.0f;
#pragma unroll
    for (int v = 0; v < V_; ++v) {
        float val = fmaxf(accL[v][threadIdx.x] + bv, 0.0f);
        yg[v * TH_ + th] = val;
        s1 += val;
        s2 += val * val;
    }
    atomicAdd(&bnsum[th], s1);
    atomicAdd(&bnsq[th], s2);
}

// ---------------------------------------------------------------------------
// BatchNorm (training stats, biased var) + residual; writes into the residual
// buffer in place (pure elementwise).
// ---------------------------------------------------------------------------
__global__ void bn_apply(const float* __restrict__ y, const float* __restrict__ bnsum,
                         const float* __restrict__ bnsq, const float* __restrict__ gamma,
                         const float* __restrict__ beta, float* __restrict__ xio)
{
    int idx = blockIdx.x * 256 + threadIdx.x;
    if (idx >= N_ * TH_) return;
    int   th  = idx & 1023, h = th & 63;
    float m   = bnsum[th] * (1.0f / N_);
    float var = bnsq[th] * (1.0f / N_) - m * m;
    float inv = rsqrtf(var + EPS_);
    xio[idx]  = gamma[h] * (y[idx] - m) * inv + beta[h] + xio[idx];
}

// ---------------------------------------------------------------------------
// Re-layout x2 [N,T,H] -> XsIn [T*B, V*H] (row = t*B+b, col = v*64+h)
// ---------------------------------------------------------------------------
__global__ void make_xsin(const float* __restrict__ x2, float* __restrict__ xsin)
{
    int idx = blockIdx.x * 256 + threadIdx.x;
    if (idx >= T_ * B_ * V_ * H_) return;
    int h    = idx & 63;
    int rest = idx >> 6;          // (t*B + b)*15 + v
    int v    = rest % V_;
    int tb   = rest / V_;
    int b    = tb & (B_ - 1);
    int t    = tb >> 11;
    xsin[idx] = x2[((size_t)(b * V_ + v) * T_ + t) * H_ + h];
}

// ---------------------------------------------------------------------------
// LSTM recurrent step: gates = xpart[t] + h @ Whh.T (biases already folded
// into xpart by the GEMM). One block per batch element, 160 threads (5 waves).
// ---------------------------------------------------------------------------
__launch_bounds__(160)
__global__ void lstm_step(const float* __restrict__ xpart, const float* __restrict__ Whh,
                          float* __restrict__ h, float* __restrict__ c, int t)
{
    const int b = blockIdx.x;
    const int j = threadIdx.x;
    __shared__ float hl[LH_];
    __shared__ float gl[4 * LH_];
    if (j < LH_) hl[j] = h[b * LH_ + j];
    __syncthreads();
    float g = xpart[((size_t)t * B_ + b) * 160 + j];
#pragma unroll
    for (int k = 0; k < LH_; ++k) g += hl[k] * Whh[j * LH_ + k];
    gl[j] = g;
    __syncthreads();
    if (j < LH_) {
        float i  = sigf(gl[j]);
        float f  = sigf(gl[LH_ + j]);
        float gg = tanhf(gl[2 * LH_ + j]);
        float o  = sigf(gl[3 * LH_ + j]);
        float cn = f * c[b * LH_ + j] + i * gg;
        c[b * LH_ + j] = cn;
        h[b * LH_ + j] = o * tanhf(cn);
    }
}

// Backward LSTM needs only its first step (zero state): c = sig(i)*tanh(g)
__global__ void lstm_back0(const float* __restrict__ xp, float* __restrict__ hb)
{
    int idx = blockIdx.x * 256 + threadIdx.x;
    if (idx >= B_ * LH_) return;
    int b = idx / LH_, k = idx % LH_;
    const float* g  = xp + (size_t)b * 160;
    float        cn = sigf(g[k]) * tanhf(g[2 * LH_ + k]);
    hb[idx] = sigf(g[3 * LH_ + k]) * tanhf(cn);
}

// ---------------------------------------------------------------------------
// Classifier: relu([hf|hb] @ Wc1.T + bc1) @ Wc2.T + bc2   (tiny)
// ---------------------------------------------------------------------------
__global__ void classifier(const float* __restrict__ hf, const float* __restrict__ hb,
                           const float* __restrict__ Wc1, const float* __restrict__ bc1,
                           const float* __restrict__ Wc2, const float* __restrict__ bc2,
                           float* __restrict__ out)
{
    int b = blockIdx.x * 256 + threadIdx.x;
    if (b >= B_) return;
    float last[80];
#pragma unroll
    for (int k = 0; k < LH_; ++k) {
        last[k]       = hf[b * LH_ + k];
        last[LH_ + k] = hb[b * LH_ + k];
    }
    float o = bc2[0];
#pragma unroll
    for (int j = 0; j < 16; ++j) {
        float s = bc1[j];
#pragma unroll
        for (int k = 0; k < 80; ++k) s += last[k] * Wc1[j * 80 + k];
        o += fmaxf(s, 0.0f) * Wc2[j];
    }
    out[b] = o;
}

// ---------------------------------------------------------------------------
extern "C" void kernel_launch(void* const* d_in, const int* in_sizes, int n_in,
                              void* d_out, int out_size, void* d_ws, size_t ws_size,
                              hipStream_t stream)
{
    (void)in_sizes; (void)n_in; (void)out_size; (void)ws_size;
    const float* data      = (const float*)d_in[0];
    const float* edge_attr = (const float*)d_in[1];
    const float* W1   = (const float*)d_in[2];
    const float* b1   = (const float*)d_in[3];
    const float* Wg1  = (const float*)d_in[4];
    const float* bg1  = (const float*)d_in[5];
    const float* gam1 = (const float*)d_in[6];
    const float* bet1 = (const float*)d_in[7];
    const float* Wg2  = (const float*)d_in[8];
    const float* bg2  = (const float*)d_in[9];
    const float* gam2 = (const float*)d_in[10];
    const float* bet2 = (const float*)d_in[11];
    const float* Wf   = (const float*)d_in[12];
    const float* bf   = (const float*)d_in[13];
    const float* WihF = (const float*)d_in[14];
    const float* WhhF = (const float*)d_in[15];
    const float* bihF = (const float*)d_in[16];
    const float* bhhF = (const float*)d_in[17];
    const float* WihB = (const float*)d_in[18];
    const float* bihB = (const float*)d_in[20];
    const float* bhhB = (const float*)d_in[21];
    const float* Wc1  = (const float*)d_in[22];
    const float* bc1  = (const float*)d_in[23];
    const float* Wc2  = (const float*)d_in[24];
    const float* bc2  = (const float*)d_in[25];
    const int*   ei   = (const int*)d_in[26];

    char*  ws  = (char*)d_ws;
    size_t off = 0;
    auto alloc = [&](size_t bytes) -> float* {
        float* p = (float*)(ws + off);
        off += (bytes + 255) & ~(size_t)255;
        return p;
    };
    float* bufX  = alloc((size_t)N_ * TH_ * 4);   // x -> x1 -> x2
    float* bufW  = alloc((size_t)N_ * TH_ * 4);   // xw / y (aliased) -> XsIn
    float* dinv  = alloc((size_t)N_ * 4);
    float* bnsum = alloc((size_t)2 * TH_ * 4);    // bnsum | bnsq
    float* bnsq  = bnsum + TH_;
    float* Xs    = alloc((size_t)T_ * B_ * H_ * 4);
    float* xpF   = alloc((size_t)T_ * B_ * 160 * 4);
    float* xpB   = alloc((size_t)B_ * 160 * 4);
    float* hF    = alloc((size_t)2 * B_ * LH_ * 4); // hF | cF
    float* cF    = hF + B_ * LH_;
    float* hB    = alloc((size_t)B_ * LH_ * 4);

    const int NE_BLK  = (N_ * TH_ + 255) / 256;   // elementwise over [N,T,H]

    // degree -> dinv
    init_deg<<<(N_ + 255) / 256, 256, 0, stream>>>(dinv);
    acc_deg<<<(E_ + 255) / 256, 256, 0, stream>>>(ei, edge_attr, dinv);
    fin_dinv<<<(N_ + 255) / 256, 256, 0, stream>>>(dinv);

    // conv_1
    conv1_kernel<<<NE_BLK, 256, 0, stream>>>(data, W1, b1, bufX);

    // --- GCN layer 1 ---
    hipMemsetAsync(bnsum, 0, 2 * TH_ * 4, stream);
    gemm_f32_wmma<4><<<(N_ * T_) / 128, 256, 0, stream>>>(
        bufX, H_, Wg1, H_, 1, nullptr, nullptr, 0, bufW, H_, N_ * T_, H_);
    gcn_agg<<<B_ * 4, 256, 0, stream>>>(bufW, ei, edge_attr, dinv, bg1, bufW, bnsum, bnsq);
    bn_apply<<<NE_BLK, 256, 0, stream>>>(bufW, bnsum, bnsq, gam1, bet1, bufX);

    // --- GCN layer 2 ---
    hipMemsetAsync(bnsum, 0, 2 * TH_ * 4, stream);
    gemm_f32_wmma<4><<<(N_ * T_) / 128, 256, 0, stream>>>(
        bufX, H_, Wg2, H_, 1, nullptr, nullptr, 0, bufW, H_, N_ * T_, H_);
    gcn_agg<<<B_ * 4, 256, 0, stream>>>(bufW, ei, edge_attr, dinv, bg2, bufW, bnsum, bnsq);
    bn_apply<<<NE_BLK, 256, 0, stream>>>(bufW, bnsum, bnsq, gam2, bet2, bufX);

    // fc1: [T*B,960] @ Wf.T -> relu -> Xs [T*B,64]
    make_xsin<<<NE_BLK, 256, 0, stream>>>(bufX, bufW);
    gemm_f32_wmma<4><<<(T_ * B_) / 128, 256, 0, stream>>>(
        bufW, V_ * H_, Wf, 1, V_ * H_, bf, nullptr, 1, Xs, H_, T_ * B_, V_ * H_);

    // LSTM input projections (all timesteps batched), biases folded in
    gemm_f32_wmma<10><<<(T_ * B_) / 128, 256, 0, stream>>>(
        Xs, H_, WihF, 1, H_, bihF, bhhF, 0, xpF, 160, T_ * B_, H_);
    gemm_f32_wmma<10><<<B_ / 128, 256, 0, stream>>>(
        Xs + (size_t)(T_ - 1) * B_ * H_, H_, WihB, 1, H_, bihB, bhhB, 0, xpB, 160, B_, H_);

    // forward recurrence
    hipMemsetAsync(hF, 0, 2 * B_ * LH_ * 4, stream);
    for (int t = 0; t < T_; ++t)
        lstm_step<<<B_, 160, 0, stream>>>(xpF, WhhF, hF, cF, t);

    // backward LSTM: single step from zero state
    lstm_back0<<<(B_ * LH_ + 255) / 256, 256, 0, stream>>>(xpB, hB);

    // classifier
    classifier<<<(B_ + 255) / 256, 256, 0, stream>>>(hF, hB, Wc1, bc1, Wc2, bc2, (float*)d_out);
}